// QuickTemplateSimpleNet_48206712930687
// MI455X (gfx1250) — compile-verified
//
#include <hip/hip_runtime.h>
#include <cstdint>

typedef float v2f __attribute__((ext_vector_type(2)));
typedef float v8f __attribute__((ext_vector_type(8)));

// ---------------------------------------------------------------- utilities
__global__ __launch_bounds__(256) void fill_f32(float* __restrict__ p, float v, int n) {
  int i = blockIdx.x * 256 + threadIdx.x;
  if (i < n) p[i] = v;
}

__global__ __launch_bounds__(256) void deg_accum(const int* __restrict__ dst,
                                                 const float* __restrict__ ew,
                                                 float* __restrict__ deg, int E) {
  int e = blockIdx.x * 256 + threadIdx.x;
  if (e < E) unsafeAtomicAdd(&deg[dst[e]], ew[e]);
}

__global__ __launch_bounds__(256) void rsqrt_inplace(float* __restrict__ p, int n) {
  int i = blockIdx.x * 256 + threadIdx.x;
  if (i < n) {
    float d = p[i];
    p[i] = (d > 0.f) ? rsqrtf(d) : 0.f;
  }
}

__global__ __launch_bounds__(256) void norm_kernel(const int* __restrict__ src,
                                                   const int* __restrict__ dst,
                                                   const float* __restrict__ ew,
                                                   const float* __restrict__ dinv,
                                                   float* __restrict__ norm, int E) {
  int e = blockIdx.x * 256 + threadIdx.x;
  if (e < E) norm[e] = dinv[src[e]] * ew[e] * dinv[dst[e]];
}

// ---------------------------------------------------------------- WMMA GEMM
// y[N,OUT] = x[N,IN] @ W[IN,OUT]   (fp32, exact, V_WMMA_F32_16X16X4_F32)
// Block: 256 threads = 8 waves; each wave owns a 16-row tile (128 rows/block).
template <int IN, int OUT>
__global__ __launch_bounds__(256) void gemm_wmma(const float* __restrict__ x,
                                                 const float* __restrict__ W,
                                                 float* __restrict__ y, int n) {
  constexpr bool USE_LDS = (IN * OUT) <= 8192;   // <=32KB -> cache W in LDS
  constexpr bool FULL    = (OUT % 16) == 0;      // no column padding needed
  __shared__ float sW[USE_LDS ? (IN * OUT) : 1];
  const float* Wp = W;
  if constexpr (USE_LDS) {
    for (int i = threadIdx.x; i < IN * OUT; i += 256) sW[i] = W[i];
    __syncthreads();
    Wp = sW;
  }

  const int lane  = threadIdx.x & 31;
  const int wave  = threadIdx.x >> 5;
  const int l15   = lane & 15;
  const int khalf = lane >> 4;          // 0: K,K+1   1: K+2,K+3
  const int row0  = blockIdx.x * 128 + wave * 16;

  if (row0 < n) {  // wave-uniform guard: EXEC stays all-ones for WMMA
    int mload = row0 + l15;
    if (mload > n - 1) mload = n - 1;   // clamp OOB rows (results discarded)
    const float* xrow = x + (size_t)mload * IN;
    const bool fulltile = (row0 + 16 <= n);  // wave-uniform

    for (int n0 = 0; n0 < OUT; n0 += 16) {
      const int colb = n0 + l15;
      // Clamp (not zero) padded columns: C[:,N] depends only on B[:,N], so
      // garbage in clamped columns never reaches stored columns. This keeps
      // the B loads unconditional -> no exec-mask toggling in the hot loop.
      const int colc = FULL ? colb : (colb < OUT ? colb : OUT - 1);
      v8f c = {};
#pragma unroll
      for (int k0 = 0; k0 < IN; k0 += 4) {
        const int krow = k0 + 2 * khalf;
        v2f a = *(const v2f*)(xrow + krow);              // A frag: K, K+1
        v2f b;
        b.x = Wp[(size_t)krow * OUT + colc];             // B row K
        b.y = Wp[(size_t)(krow + 1) * OUT + colc];       // B row K+1
        c = __builtin_amdgcn_wmma_f32_16x16x4_f32(
            /*neg_a=*/false, a, /*neg_b=*/false, b,
            /*c_mod=*/(short)0, c, /*reuse_a=*/false, /*reuse_b=*/false);
      }
      const int rowoff = khalf * 8;      // C/D: lanes 16-31 hold M = r+8
      if (FULL && fulltile) {            // fast path: unguarded stores
#pragma unroll
        for (int r = 0; r < 8; ++r)
          y[(size_t)(row0 + rowoff + r) * OUT + colb] = c[r];
      } else {
        const bool colok = FULL || (colb < OUT);
#pragma unroll
        for (int r = 0; r < 8; ++r) {
          int row = row0 + rowoff + r;
          if (row < n && colok) y[(size_t)row * OUT + colb] = c[r];
        }
      }
    }
  }
}

// ------------------------------------------------- self-loop + bias (init)
template <int OUT>
__global__ __launch_bounds__(256) void self_bias(const float* __restrict__ xw,
                                                 const float* __restrict__ dinv,
                                                 const float* __restrict__ b,
                                                 float* __restrict__ out, int n) {
  int tid = blockIdx.x * 256 + threadIdx.x;
  if (tid >= n * OUT) return;
  int i = tid / OUT, c = tid - i * OUT;
  float dv = dinv[i];
  out[tid] = xw[tid] * (dv * dv) + b[c];
}

// ---------------------------------------------------- edge scatter (atomic)
template <int OUT>
__global__ __launch_bounds__(256) void edge_scatter(const float* __restrict__ xw,
                                                    const int* __restrict__ src,
                                                    const int* __restrict__ dst,
                                                    const float* __restrict__ norm,
                                                    float* __restrict__ out, int E) {
  int tid = blockIdx.x * 256 + threadIdx.x;
  if (tid >= E * OUT) return;
  int e = tid / OUT, c = tid - e * OUT;
  float v = xw[(size_t)src[e] * OUT + c] * norm[e];
  unsafeAtomicAdd(&out[(size_t)dst[e] * OUT + c], v);
}

// ---------------------------------------------------------------- ELU
__global__ __launch_bounds__(256) void elu_inplace(float* __restrict__ p, int n) {
  int i = blockIdx.x * 256 + threadIdx.x;
  if (i < n) {
    float v = p[i];
    p[i] = (v > 0.f) ? v : expm1f(v);
  }
}

// ---------------------------------------------------------------- launcher
static inline int cdiv(long long a, long long b) { return (int)((a + b - 1) / b); }

template <int IN, int OUT>
static void run_layer(const float* xin, float* xw, float* xout,
                      const float* W, const float* b,
                      const int* src, const int* dst,
                      const float* dinv, const float* norm,
                      int N, int E, bool act, hipStream_t stream) {
  gemm_wmma<IN, OUT><<<cdiv(N, 128), 256, 0, stream>>>(xin, W, xw, N);
  self_bias<OUT><<<cdiv((long long)N * OUT, 256), 256, 0, stream>>>(xw, dinv, b, xout, N);
  edge_scatter<OUT><<<cdiv((long long)E * OUT, 256), 256, 0, stream>>>(xw, src, dst, norm, xout, E);
  if (act) elu_inplace<<<cdiv((long long)N * OUT, 256), 256, 0, stream>>>(xout, N * OUT);
}

extern "C" void kernel_launch(void* const* d_in, const int* in_sizes, int n_in,
                              void* d_out, int out_size, void* d_ws, size_t ws_size,
                              hipStream_t stream) {
  const float* x   = (const float*)d_in[0];
  const int*   ei  = (const int*)d_in[1];    // [2, E] flattened (int32: JAX x64 off)
  const float* ew  = (const float*)d_in[2];
  const float* W1 = (const float*)d_in[3];  const float* b1 = (const float*)d_in[4];
  const float* W2 = (const float*)d_in[5];  const float* b2 = (const float*)d_in[6];
  const float* W3 = (const float*)d_in[7];  const float* b3 = (const float*)d_in[8];
  const float* W4 = (const float*)d_in[9];  const float* b4 = (const float*)d_in[10];
  const float* W5 = (const float*)d_in[11]; const float* b5 = (const float*)d_in[12];
  const float* W6 = (const float*)d_in[13]; const float* b6 = (const float*)d_in[14];

  const int N = in_sizes[0] / 32;
  const int E = in_sizes[2];
  const int* src = ei;
  const int* dst = ei + E;

  float* dinv = (float*)d_ws;           // N
  float* norm = dinv + N;               // E
  float* buf0 = norm + E;               // N*128  (layer activations)
  float* buf1 = buf0 + (size_t)N * 128; // N*128  (xw scratch)
  float* outp = (float*)d_out;

  // GCN normalization: deg = 1 (self loop) + sum_e ew[dst]; dinv = rsqrt(deg)
  fill_f32<<<cdiv(N, 256), 256, 0, stream>>>(dinv, 1.0f, N);
  deg_accum<<<cdiv(E, 256), 256, 0, stream>>>(dst, ew, dinv, E);
  rsqrt_inplace<<<cdiv(N, 256), 256, 0, stream>>>(dinv, N);
  norm_kernel<<<cdiv(E, 256), 256, 0, stream>>>(src, dst, ew, dinv, norm, E);

  // conv1..conv6 (ELU after 1,2,3,5)
  run_layer<32,   4>(x,    buf1, buf0, W1, b1, src, dst, dinv, norm, N, E, true,  stream);
  run_layer<4,   16>(buf0, buf1, buf0, W2, b2, src, dst, dinv, norm, N, E, true,  stream);
  run_layer<16,  32>(buf0, buf1, buf0, W3, b3, src, dst, dinv, norm, N, E, true,  stream);
  run_layer<32,  64>(buf0, buf1, buf0, W4, b4, src, dst, dinv, norm, N, E, false, stream);
  run_layer<64, 128>(buf0, buf1, buf0, W5, b5, src, dst, dinv, norm, N, E, true,  stream);
  run_layer<128,128>(buf0, buf1, outp, W6, b6, src, dst, dinv, norm, N, E, false, stream);
}